// DiffMC_44848048505120
// MI455X (gfx1250) — compile-verified
//
#include <hip/hip_runtime.h>

typedef float v2f __attribute__((ext_vector_type(2)));
typedef float v8f __attribute__((ext_vector_type(8)));

#define NC    65              // cells per axis (padded grid 66 -> 65 cells)
#define NCELL (65 * 65 * 65)  // 274625
#define NEDGE 19

// Unique cell-edge endpoint corners, in discovery order (matches reference T2C build)
__constant__ int c_ea[NEDGE] = {0,0,0,1,1,3,0,2,2,0,2,6,0,4,4,0,4,5,1};
__constant__ int c_eb[NEDGE] = {1,3,7,3,7,7,2,3,7,6,6,7,4,6,7,5,5,7,5};

__constant__ int c_t2c[6][6] = {
    {0,1,2,3,4,5}, {1,6,2,7,5,8}, {6,9,2,10,8,11},
    {9,12,2,13,11,14}, {12,15,2,16,14,17}, {15,0,2,18,17,4}};

__constant__ int c_tets[6][4] = {
    {0,1,3,7},{0,3,2,7},{0,2,6,7},{0,6,4,7},{0,4,5,7},{0,5,1,7}};

__constant__ int c_mt[16][6] = {
    {-1,-1,-1,-1,-1,-1},{0,1,2,-1,-1,-1},{0,3,4,-1,-1,-1},{1,2,4,1,4,3},
    {1,3,5,-1,-1,-1},{0,2,5,0,5,3},{0,1,5,0,5,4},{2,4,5,-1,-1,-1},
    {2,4,5,-1,-1,-1},{0,1,5,0,5,4},{0,3,5,0,5,2},{1,3,5,-1,-1,-1},
    {1,3,4,1,4,2},{0,3,4,-1,-1,-1},{0,1,2,-1,-1,-1},{-1,-1,-1,-1,-1,-1}};

// Load one feature value of a padded-corner (zero outside the real 64^3 grid / n>=8)
__device__ __forceinline__ float load_feat(const float* __restrict__ feat,
                                           int i, int j, int k, int c, int n) {
    int ci = i + (c & 1) - 1;
    int cj = j + ((c >> 1) & 1) - 1;
    int ck = k + ((c >> 2) & 1) - 1;
    if ((unsigned)ci >= 64u || (unsigned)cj >= 64u || (unsigned)ck >= 64u || n >= 8)
        return 0.0f;
    return feat[(((ci * 64) + cj) * 64 + ck) * 8 + n];
}

// corner weight for the W matrix row (edge): (1-t) at ea, t at eb, gated by cross
__device__ __forceinline__ float wgt(int c, float t, int cr, int ea, int eb) {
    float w = (c == ea) ? (1.0f - t) : ((c == eb) ? t : 0.0f);
    return cr ? w : 0.0f;
}

__global__ __launch_bounds__(256)
void diffmc_kernel(const float* __restrict__ sdf, const float* __restrict__ feat,
                   float* __restrict__ outV, float* __restrict__ outF,
                   int* __restrict__ outT) {
    const int lane = threadIdx.x & 31;
    const int cell = blockIdx.x * 8 + (threadIdx.x >> 5);
    if (cell >= NCELL) return;  // wave-uniform: EXEC stays all-ones for WMMA

    const int i = cell / (NC * NC);
    const int rem = cell - i * NC * NC;
    const int j = rem / NC;
    const int k = rem - j * NC;

    // ---- corner SDF values: lane c<8 loads padded corner c (pad value = iso+1 = 1.0)
    float sraw = 1.0f;
    {
        int c = lane & 7;
        int ci = i + (c & 1) - 1;
        int cj = j + ((c >> 1) & 1) - 1;
        int ck = k + ((c >> 2) & 1) - 1;
        if (lane < 8 && (unsigned)ci < 64u && (unsigned)cj < 64u && (unsigned)ck < 64u)
            sraw = sdf[(ci * 64 + cj) * 64 + ck];
    }

    // inside bitmask over 8 corners (broadcast via shuffle)
    unsigned imask = 0;
#pragma unroll
    for (int c = 0; c < 8; ++c) {
        float sc = __shfl(sraw, c);
        imask |= (sc < 0.0f ? 1u : 0u) << c;
    }

    // ---- per-edge crossing + t (lane e handles edge e, e<19)
    const int e  = (lane < NEDGE) ? lane : NEDGE - 1;
    const int ea = c_ea[e], eb = c_eb[e];
    const float sa = __shfl(sraw, ea);
    const float sb = __shfl(sraw, eb);
    const int cross_l = (((imask >> ea) ^ (imask >> eb)) & 1) ? 1 : 0;
    const float t_l = cross_l ? (0.0f - sa) / (sb - sa) : 0.0f;

    // ---- verts: lane e writes its 3 floats
    if (lane < NEDGE) {
        float ax = (float)(i + (ea & 1)),        ay = (float)(j + ((ea >> 1) & 1)),
              az = (float)(k + ((ea >> 2) & 1));
        float bx = (float)(i + (eb & 1)),        by = (float)(j + ((eb >> 1) & 1)),
              bz = (float)(k + ((eb >> 2) & 1));
        float vx = ax + t_l * (bx - ax);
        float vy = ay + t_l * (by - ay);
        float vz = az + t_l * (bz - az);
        if (!cross_l) { vx = 0.0f; vy = 0.0f; vz = 0.0f; }
        const float inv = 1.0f / 63.0f;
        int base = (cell * NEDGE + lane) * 3;
        outV[base + 0] = (vx - 1.0f) * inv;
        outV[base + 1] = (vy - 1.0f) * inv;
        outV[base + 2] = (vz - 1.0f) * inv;
    }

    // ======== feature interpolation as WMMA:  D(16x16) = W(16x8) x CF(8x16) ========
    // B matrix (corner features), shared by both M-tiles.
    // f32 16x16x4 B layout: col N = lane&15; b.x: K = (lane<16?0:2), b.y: K+1.
    const int n  = lane & 15;
    const int kb = (lane < 16) ? 0 : 2;
    v2f b0, b1;
    b0.x = load_feat(feat, i, j, k, kb,     n);
    b0.y = load_feat(feat, i, j, k, kb + 1, n);
    b1.x = load_feat(feat, i, j, k, kb + 4, n);
    b1.y = load_feat(feat, i, j, k, kb + 5, n);

    const int row = lane & 15;
    const int ka  = (lane < 16) ? 0 : 2;  // A layout: a.x: K=ka, a.y: K=ka+1

    // ---- tile 0: rows = edges 0..15
    {
        float t0 = __shfl(t_l, row);
        int  cr0 = __shfl(cross_l, row);
        int  ea0 = c_ea[row], eb0 = c_eb[row];
        v2f a0, a1;
        a0.x = wgt(ka,     t0, cr0, ea0, eb0);
        a0.y = wgt(ka + 1, t0, cr0, ea0, eb0);
        a1.x = wgt(ka + 4, t0, cr0, ea0, eb0);
        a1.y = wgt(ka + 5, t0, cr0, ea0, eb0);
        v8f acc = {};
        acc = __builtin_amdgcn_wmma_f32_16x16x4_f32(false, a0, false, b0, (short)0, acc, false, false);
        acc = __builtin_amdgcn_wmma_f32_16x16x4_f32(false, a1, false, b1, (short)0, acc, false, false);
        if (n < 8) {
#pragma unroll
            for (int v = 0; v < 8; ++v) {
                int M = v + ((lane < 16) ? 0 : 8);   // edge id 0..15
                outF[(cell * NEDGE + M) * 8 + n] = acc[v];
            }
        }
    }

    // ---- tile 1: rows 0..2 = edges 16..18 (rest zero-weight)
    {
        bool vr = (row < 3);
        int  s1 = vr ? (16 + row) : (NEDGE - 1);
        float t1 = __shfl(t_l, s1);
        int  cr1 = __shfl(cross_l, s1);
        if (!vr) { t1 = 0.0f; cr1 = 0; }
        int ea1 = c_ea[s1], eb1 = c_eb[s1];
        v2f a0, a1;
        a0.x = wgt(ka,     t1, cr1, ea1, eb1);
        a0.y = wgt(ka + 1, t1, cr1, ea1, eb1);
        a1.x = wgt(ka + 4, t1, cr1, ea1, eb1);
        a1.y = wgt(ka + 5, t1, cr1, ea1, eb1);
        v8f acc = {};
        acc = __builtin_amdgcn_wmma_f32_16x16x4_f32(false, a0, false, b0, (short)0, acc, false, false);
        acc = __builtin_amdgcn_wmma_f32_16x16x4_f32(false, a1, false, b1, (short)0, acc, false, false);
        if (n < 8 && lane < 16) {
#pragma unroll
            for (int v = 0; v < 3; ++v)              // edges 16,17,18
                outF[(cell * NEDGE + 16 + v) * 8 + n] = acc[v];
        }
    }

    // ---- tris: 36 int entries per cell
#pragma unroll
    for (int pass = 0; pass < 2; ++pass) {
        int idx = lane + pass * 32;
        if (idx < 36) {
            int tet = idx / 6, jj = idx - tet * 6;
            int cas = ((imask >> c_tets[tet][0]) & 1)
                    | (((imask >> c_tets[tet][1]) & 1) << 1)
                    | (((imask >> c_tets[tet][2]) & 1) << 2)
                    | (((imask >> c_tets[tet][3]) & 1) << 3);
            int lt  = c_mt[cas][jj];
            int gid = (lt >= 0) ? (cell * NEDGE + c_t2c[tet][lt]) : -1;
            outT[cell * 36 + idx] = gid;
        }
    }
}

extern "C" void kernel_launch(void* const* d_in, const int* in_sizes, int n_in,
                              void* d_out, int out_size, void* d_ws, size_t ws_size,
                              hipStream_t stream) {
    const float* sdf  = (const float*)d_in[0];
    const float* feat = (const float*)d_in[1];
    float* outV = (float*)d_out;
    float* outF = outV + (size_t)NCELL * NEDGE * 3;
    int*   outT = (int*)(outF + (size_t)NCELL * NEDGE * 8);
    int blocks = (NCELL + 7) / 8;  // 8 waves (cells) per 256-thread block
    diffmc_kernel<<<blocks, 256, 0, stream>>>(sdf, feat, outV, outF, outT);
}